// Attn_69776038691596
// MI455X (gfx1250) — compile-verified
//
#include <hip/hip_runtime.h>
#include <math.h>

typedef float v2f __attribute__((ext_vector_type(2)));
typedef float v4f __attribute__((ext_vector_type(4)));
typedef float v8f __attribute__((ext_vector_type(8)));

#define BATCH 64
#define SEQ   2048
#define HID   1024

// ---------------------------------------------------------------------------
// Kernel 1: v = h @ W   (M=64, N=1024, K=1024), fp32 WMMA 16x16x4.
// One wave (32 threads) per 16x16 output tile. grid = 4 Mtiles * 64 Ntiles.
// A layout (16x4 f32): lane<16 -> {A[m=lane][k0], A[m=lane][k0+1]},
//                      lane>=16 -> {A[m=lane-16][k0+2], A[m=lane-16][k0+3]}
// B layout (4x16 f32): mirrored: vgpr0 = B[k0+2*half][n0+lm], vgpr1 = next k.
// D layout (16x16 f32): vgpr j -> M = m0 + j + 8*half, N = n0 + lm.
// ---------------------------------------------------------------------------
__global__ __launch_bounds__(32) void gemm_v_kernel(const float* __restrict__ hid,
                                                    const float* __restrict__ W,
                                                    float* __restrict__ v)
{
    const int lane = threadIdx.x;        // 0..31
    const int half = lane >> 4;          // 0 or 1
    const int lm   = lane & 15;
    const int nt   = blockIdx.x & 63;    // 64 N tiles
    const int mt   = blockIdx.x >> 6;    // 4  M tiles
    const int m0   = mt * 16;
    const int n0   = nt * 16;

    const float* __restrict__ arow = hid + (size_t)(m0 + lm) * HID;
    const float* __restrict__ bcol = W + n0 + lm;

    v8f acc = {};
#pragma unroll 4
    for (int k0 = 0; k0 < HID; k0 += 4) {
        const int ka = k0 + 2 * half;
        v2f a = *(const v2f*)(arow + ka);                 // A[m][ka], A[m][ka+1]
        v2f b;
        b.x = bcol[(size_t)ka * HID];                     // W[ka][n0+lm]
        b.y = bcol[(size_t)(ka + 1) * HID];               // W[ka+1][n0+lm]
        acc = __builtin_amdgcn_wmma_f32_16x16x4_f32(
                  /*neg_a=*/false, a, /*neg_b=*/false, b,
                  /*c_mod=*/(short)0, acc,
                  /*reuse_a=*/false, /*reuse_b=*/false);
    }

#pragma unroll
    for (int j = 0; j < 8; ++j) {
        const int m = m0 + j + 8 * half;
        v[(size_t)m * HID + n0 + lm] = acc[j];
    }
}

// ---------------------------------------------------------------------------
// Kernel 2: c[b] = dot(h[b,:], bias)   (64 blocks x 256 threads, LDS reduce)
// ---------------------------------------------------------------------------
__global__ __launch_bounds__(256) void bias_dot_kernel(const float* __restrict__ hid,
                                                       const float* __restrict__ bias,
                                                       float* __restrict__ c)
{
    __shared__ float red[256];
    const int b = blockIdx.x;
    const int t = threadIdx.x;
    float p = 0.f;
#pragma unroll
    for (int i = 0; i < HID / 256; ++i) {
        const int k = t + i * 256;
        p += hid[(size_t)b * HID + k] * bias[k];
    }
    red[t] = p;
    __syncthreads();
#pragma unroll
    for (int s = 128; s > 0; s >>= 1) {
        if (t < s) red[t] += red[t + s];
        __syncthreads();
    }
    if (t == 0) c[b] = red[0];
}

// ---------------------------------------------------------------------------
// Kernel 3: E[b,s] = enc[b,s,:] . v[b,:] + c[b]     (HBM-bound: 512 MB read)
// Block = 256 threads (8 waves), one batch b per block, 64 s-rows per block
// (8 rows per wave). v[b,:] staged once in LDS (4 KB).
// encoder_output is a 512 MB one-shot stream (>> 192 MB L2) -> load it with
// non-temporal hints so it doesn't evict the small reused tensors.
// ---------------------------------------------------------------------------
__global__ __launch_bounds__(256) void energies_kernel(const float* __restrict__ enc,
                                                       const float* __restrict__ v,
                                                       const float* __restrict__ c,
                                                       float* __restrict__ E)
{
    __shared__ float vs[HID];            // 4 KB
    const int b = blockIdx.y;
    const int t = threadIdx.x;

    // stage v[b,:] -> LDS (256 threads x 4 floats = 1024 floats)
    ((v4f*)vs)[t] = ((const v4f*)(v + (size_t)b * HID))[t];
    __syncthreads();

    const int wave   = t >> 5;
    const int lane   = t & 31;
    const int s_base = blockIdx.x * 64 + wave * 8;
    const float cb   = c[b];

#pragma unroll
    for (int r = 0; r < 8; ++r) {
        const int s = s_base + r;
        const v4f* __restrict__ row =
            (const v4f*)(enc + ((size_t)b * SEQ + (size_t)s) * HID);
        float acc = 0.f;
#pragma unroll
        for (int i = 0; i < HID / 128; ++i) {            // 8 iters of 128 floats
            const v4f e = __builtin_nontemporal_load(&row[lane + i * 32]);
            const v4f w = ((const v4f*)vs)[lane + i * 32];
            acc += e.x * w.x + e.y * w.y + e.z * w.z + e.w * w.w;
        }
        // wave32 butterfly reduction
#pragma unroll
        for (int off = 16; off > 0; off >>= 1)
            acc += __shfl_xor(acc, off, 32);
        if (lane == 0) E[(size_t)b * SEQ + s] = acc + cb;
    }
}

// ---------------------------------------------------------------------------
// Kernel 4: softmax over axis 0 (batch) of E (64 values per column s),
// done in-place on d_out. One thread per s; 64 vals fully unrolled in VGPRs.
// ---------------------------------------------------------------------------
__global__ __launch_bounds__(256) void softmax_b_kernel(float* __restrict__ E)
{
    const int s = blockIdx.x * 256 + threadIdx.x;        // 0..2047
    float vals[BATCH];
    float mx = -INFINITY;
#pragma unroll
    for (int b = 0; b < BATCH; ++b) {
        vals[b] = E[(size_t)b * SEQ + s];
        mx = fmaxf(mx, vals[b]);
    }
    float sum = 0.f;
#pragma unroll
    for (int b = 0; b < BATCH; ++b) {
        vals[b] = __expf(vals[b] - mx);
        sum += vals[b];
    }
    const float inv = 1.0f / sum;
#pragma unroll
    for (int b = 0; b < BATCH; ++b)
        E[(size_t)b * SEQ + s] = vals[b] * inv;
}

// ---------------------------------------------------------------------------
extern "C" void kernel_launch(void* const* d_in, const int* in_sizes, int n_in,
                              void* d_out, int out_size, void* d_ws, size_t ws_size,
                              hipStream_t stream)
{
    const float* hid  = (const float*)d_in[0];   // (1, 64, 1024)
    const float* enc  = (const float*)d_in[1];   // (64, 2048, 1024)
    const float* W    = (const float*)d_in[2];   // (1024, 1024)
    const float* bias = (const float*)d_in[3];   // (1024,)

    float* E = (float*)d_out;                    // energies live in d_out (B,S)

    // workspace: v (64*1024 f32 = 256 KB) + c (64 f32)
    float* v = (float*)d_ws;
    float* c = (float*)((char*)d_ws + (size_t)BATCH * HID * sizeof(float));

    // 1) v = h @ W via fp32 WMMA (4 Mtiles * 64 Ntiles, 1 wave each)
    gemm_v_kernel<<<dim3(4 * 64), dim3(32), 0, stream>>>(hid, W, v);

    // 2) c[b] = h[b] . bias
    bias_dot_kernel<<<dim3(BATCH), dim3(256), 0, stream>>>(hid, bias, c);

    // 3) energies (bandwidth-bound pass over encoder_output, NT loads)
    energies_kernel<<<dim3(SEQ / 64, BATCH), dim3(256), 0, stream>>>(enc, v, c, E);

    // 4) softmax over batch axis, in place
    softmax_b_kernel<<<dim3(SEQ / 256), dim3(256), 0, stream>>>(E);
}